// NRE_DCT_51135880626463
// MI455X (gfx1250) — compile-verified
//
#include <hip/hip_runtime.h>
#include <math.h>

// ---------------------------------------------------------------------------
// DCT-masked reconstruction for MI455X (gfx1250, wave32, WMMA f32 16x16x4).
//
//   out = [ residual (32*1024*3*256) | recon (same) | Mf (768) ]  fp32
//
// Kernel 1 (1 block): builds the 16x16 DCT matrix D into workspace and runs
//   the tiny frequency-MLP -> sigmoid mask Mf (768 floats) into d_out tail.
// Kernel 2 (12288 blocks x 256 thr = 8 waves): one wave per (b,n,c) patch:
//   T1 = D*P ; Dcoeff = T1*D^T ; mask in regs ; T2 = D^T*Dm ; recon = T2*D ;
//   residual = P - recon.  All four 16x16x16 GEMMs via V_WMMA_F32_16X16X4_F32.
//
// Stage-to-stage LDS reuse is wave-private: CDNA5 LDS ops from one wave are
// executed in order (DScnt), so no block barriers are needed between stages —
// only a compiler scheduling fence. This lets the 8 waves per block drift and
// overlap WMMA with LDS/global traffic instead of lock-stepping on barriers.
// ---------------------------------------------------------------------------

typedef __attribute__((ext_vector_type(2))) float v2f;
typedef __attribute__((ext_vector_type(8))) float v8f;

constexpr int    kB      = 32;
constexpr int    kC      = 3;
constexpr int    kNP     = 1024;                 // (512/16)^2
constexpr int    kUnits  = kB * kNP * kC;        // 98304 patch-channels
constexpr size_t kElems  = (size_t)kUnits * 256; // 25165824

// Compile-time scheduling fence: no instruction may move across it.
// Emits nothing; hardware per-wave LDS ordering provides the actual RAW/WAR
// guarantees for the wave-private tiles.
__device__ __forceinline__ void stage_fence() {
  __builtin_amdgcn_sched_barrier(0);
}

__device__ __forceinline__ v8f wmma_f32_k4(v2f a, v2f b, v8f c) {
  // D = A(16x4,f32) * B(4x16,f32) + C(16x16,f32)   -> v_wmma_f32_16x16x4_f32
  return __builtin_amdgcn_wmma_f32_16x16x4_f32(
      /*neg_a=*/false, a, /*neg_b=*/false, b,
      /*c_mod=*/(short)0, c, /*reuse_a=*/false, /*reuse_b=*/false);
}

// C = Frag(A) * B,  B gathered from row-major 16x16 LDS tile: b = B[K][N]
__device__ __forceinline__ v8f gemm_Afrag(const v2f* F, const float* B,
                                          int lo, int hi) {
  v8f acc = {};
#pragma unroll
  for (int ks = 0; ks < 4; ++ks) {
    const int ka = ks * 4 + 2 * hi;
    v2f b;
    b.x = B[ka * 16 + lo];
    b.y = B[(ka + 1) * 16 + lo];
    acc = wmma_f32_k4(F[ks], b, acc);
  }
  return acc;
}

// C = A * Frag(B),  A gathered from row-major 16x16 LDS tile: a = A[M][K]
__device__ __forceinline__ v8f gemm_Bfrag(const float* A, const v2f* F,
                                          int lo, int hi) {
  v8f acc = {};
#pragma unroll
  for (int ks = 0; ks < 4; ++ks) {
    const int ka = ks * 4 + 2 * hi;
    v2f a;
    a.x = A[lo * 16 + ka];
    a.y = A[lo * 16 + ka + 1];
    acc = wmma_f32_k4(a, F[ks], acc);
  }
  return acc;
}

// C/D tile layout: VGPR r, lane -> (M = r + 8*hi, N = lo)
__device__ __forceinline__ void scatter16(float* T, v8f acc, int lo, int hi) {
#pragma unroll
  for (int r = 0; r < 8; ++r) T[(r + 8 * hi) * 16 + lo] = acc[r];
}

// ---------------------------------------------------------------------------
// Prologue: DCT matrix + MLP mask (single block, 256 threads; thread p = coord)
// ---------------------------------------------------------------------------
__global__ __launch_bounds__(256) void dct_mask_prologue(
    const float* __restrict__ W1, const float* __restrict__ b1,
    const float* __restrict__ W2, const float* __restrict__ b2,
    const float* __restrict__ W3, const float* __restrict__ b3,
    float* __restrict__ Dws, float* __restrict__ MfOut) {
  __shared__ float hbar[64];
  const int tid = threadIdx.x;
  if (tid < 64) hbar[tid] = 0.0f;

  // D[i][k] = cos(pi*(2i+1)*k/32) * sqrt(2/16), row 0 scaled by 1/sqrt(2)
  {
    const int i = tid >> 4, k = tid & 15;
    float v = cosf(3.14159265358979323846f * (float)((2 * i + 1) * k) *
                   (1.0f / 32.0f)) *
              0.35355339059327379f;
    if (i == 0) v *= 0.70710678118654752f;
    Dws[tid] = v;
  }
  __syncthreads();

  // coords[p] = (p/16, p%16) / 15
  const float gx = (float)(tid >> 4) * (1.0f / 15.0f);
  const float gy = (float)(tid & 15) * (1.0f / 15.0f);

  float h1[64];
#pragma unroll
  for (int h = 0; h < 64; ++h) {
    float a = fmaf(gx, W1[h * 2 + 0], fmaf(gy, W1[h * 2 + 1], b1[h]));
    h1[h] = a > 0.0f ? a : 0.0f;
  }
  // mean over coords folds into hbar[k] = mean_p h2[p][k]
  for (int h = 0; h < 64; ++h) {
    float a = b2[h];
#pragma unroll
    for (int k = 0; k < 64; ++k) a = fmaf(h1[k], W2[h * 64 + k], a);
    a = a > 0.0f ? a : 0.0f;
    atomicAdd(&hbar[h], a * (1.0f / 256.0f));
  }
  __syncthreads();

  // Mf[o] = sigmoid(hbar . W3[o] + b3[o]),  o = 0..767
#pragma unroll
  for (int oo = 0; oo < 3; ++oo) {
    const int o = tid + oo * 256;
    float a = b3[o];
    for (int k = 0; k < 64; ++k) a = fmaf(hbar[k], W3[o * 64 + k], a);
    MfOut[o] = 1.0f / (1.0f + expf(-a));
  }
}

// ---------------------------------------------------------------------------
// Main kernel: one wave32 per (b,n,c) patch-channel, 8 waves per block.
// ---------------------------------------------------------------------------
__global__ __launch_bounds__(256) void dct_mask_recon(
    const float* __restrict__ x, const float* __restrict__ Dws,
    const float* __restrict__ Mf, float* __restrict__ out_res,
    float* __restrict__ out_rec) {
  __shared__ float Dlds[256];
  __shared__ float Mlds[768];
  __shared__ __align__(16) float Pbuf[8][256];
  __shared__ __align__(16) float Tbuf[8][256];

  const int tid  = threadIdx.x;
  const int wave = tid >> 5;
  const int lane = tid & 31;
  const int lo   = lane & 15;   // N / M row-in-halfwave index
  const int hi   = lane >> 4;   // halfwave select

  Dlds[tid]       = Dws[tid];
  Mlds[tid]       = Mf[tid];
  Mlds[tid + 256] = Mf[tid + 256];
  Mlds[tid + 512] = Mf[tid + 512];
  __syncthreads();   // only true cross-wave dependency (block-shared D, M)

  // Invariant D-side fragments:
  //   F1 = A-frag(D)   == B-frag(D^T)  : element D[lo][K]
  //   F2 = A-frag(D^T) == B-frag(D)    : element D[K][lo]
  v2f F1[4], F2[4];
#pragma unroll
  for (int ks = 0; ks < 4; ++ks) {
    const int ka = ks * 4 + 2 * hi;
    F1[ks].x = Dlds[lo * 16 + ka];
    F1[ks].y = Dlds[lo * 16 + ka + 1];
    F2[ks].x = Dlds[ka * 16 + lo];
    F2[ks].y = Dlds[(ka + 1) * 16 + lo];
  }

  const int unit = blockIdx.x * 8 + wave;   // (b*1024+n)*3 + c
  const int c    = unit % 3;
  const int t    = unit / 3;
  const int n    = t & 1023;
  const int b    = t >> 10;
  const int ph   = n >> 5, pw = n & 31;

  const float* xp = x + (size_t)(b * 3 + c) * 262144 +
                    (size_t)(ph * 16) * 512 + (size_t)(pw * 16);

  float* Pl = Pbuf[wave];   // wave-private 16x16 tile
  float* Tl = Tbuf[wave];   // wave-private 16x16 tile

  // Load 16x16 patch: lane -> row rr = lane/2, cols cc..cc+7  (2x b128)
  const int rr = lane >> 1;
  const int cc = (lane & 1) * 8;
  const float4 p0 = *(const float4*)(xp + rr * 512 + cc);
  const float4 p1 = *(const float4*)(xp + rr * 512 + cc + 4);
  *(float4*)(Pl + rr * 16 + cc)     = p0;
  *(float4*)(Pl + rr * 16 + cc + 4) = p1;
  stage_fence();   // per-wave LDS RAW: hardware in-order DS, no barrier needed

  // T1 = D * P
  v8f acc = gemm_Afrag(F1, Pl, lo, hi);
  scatter16(Tl, acc, lo, hi);
  stage_fence();

  // Dcoeff = T1 * D^T, then mask in registers
  acc = gemm_Bfrag(Tl, F1, lo, hi);
  {
    const float* Mc = Mlds + c * 256;
#pragma unroll
    for (int r = 0; r < 8; ++r) acc[r] *= Mc[(r + 8 * hi) * 16 + lo];
  }
  stage_fence();
  scatter16(Tl, acc, lo, hi);
  stage_fence();

  // T2 = D^T * Dmasked
  acc = gemm_Afrag(F2, Tl, lo, hi);
  stage_fence();
  scatter16(Tl, acc, lo, hi);
  stage_fence();

  // recon = T2 * D
  acc = gemm_Bfrag(Tl, F2, lo, hi);
  stage_fence();
  scatter16(Tl, acc, lo, hi);
  stage_fence();

  // Stream out recon and residual = P - recon as float4s (1KB contiguous
  // per store pair across the wave; each b128 store is 16B/lane at 32B stride).
  const size_t obase = (size_t)unit * 256 + (size_t)(rr * 16 + cc);
  const float4 rc0 = *(const float4*)(Tl + rr * 16 + cc);
  const float4 rc1 = *(const float4*)(Tl + rr * 16 + cc + 4);
  float4 rs0, rs1;
  rs0.x = p0.x - rc0.x; rs0.y = p0.y - rc0.y;
  rs0.z = p0.z - rc0.z; rs0.w = p0.w - rc0.w;
  rs1.x = p1.x - rc1.x; rs1.y = p1.y - rc1.y;
  rs1.z = p1.z - rc1.z; rs1.w = p1.w - rc1.w;
  *(float4*)(out_rec + obase)     = rc0;
  *(float4*)(out_rec + obase + 4) = rc1;
  *(float4*)(out_res + obase)     = rs0;
  *(float4*)(out_res + obase + 4) = rs1;
}

extern "C" void kernel_launch(void* const* d_in, const int* in_sizes, int n_in,
                              void* d_out, int out_size, void* d_ws,
                              size_t ws_size, hipStream_t stream) {
  (void)in_sizes; (void)n_in; (void)out_size; (void)ws_size;
  const float* x  = (const float*)d_in[0];
  const float* W1 = (const float*)d_in[1];
  const float* b1 = (const float*)d_in[2];
  const float* W2 = (const float*)d_in[3];
  const float* b2 = (const float*)d_in[4];
  const float* W3 = (const float*)d_in[5];
  const float* b3 = (const float*)d_in[6];

  float* out     = (float*)d_out;
  float* out_res = out;                       // residual
  float* out_rec = out + kElems;              // recon
  float* out_mf  = out + 2 * kElems;          // Mf (768)
  float* Dws     = (float*)d_ws;              // 256 floats: DCT matrix

  dct_mask_prologue<<<1, 256, 0, stream>>>(W1, b1, W2, b2, W3, b3, Dws, out_mf);
  dct_mask_recon<<<kUnits / 8, 256, 0, stream>>>(x, Dws, out_mf, out_res,
                                                 out_rec);
}